// Network_5952824672836
// MI455X (gfx1250) — compile-verified
//
#include <hip/hip_runtime.h>

#define N 4096
#define KSPLIT 16
#define KCHUNK (N / KSPLIT)        // 256 rows per split
#define WAVES_PER_BLOCK 8
#define COLS_PER_WAVE 16

typedef __attribute__((ext_vector_type(2))) float v2f;
typedef __attribute__((ext_vector_type(4))) float v4f;
typedef __attribute__((ext_vector_type(8))) float v8f;

// ---------------------------------------------------------------------------
// Kernel 1: split-K GEMV partials via V_WMMA_F32_16X16X4_F32.
// Each wave owns 16 output columns; loops kb over its K-chunk in steps of 4.
//   A (16x4, MxK): A[m][k] = M[kb+k][colbase+m],  M = w + alpha*hebb
//     ISA layout: lanes 0-15 -> M=lane, VGPR0=K0, VGPR1=K1
//                 lanes 16-31 -> M=lane-16, VGPR0=K2, VGPR1=K3
//   B (4x16, KxN): only column 0 nonzero, B[k][0] = yin[kb+k]
//     -> lane0: {yin[kb], yin[kb+1]}, lane16: {yin[kb+2], yin[kb+3]}, else 0
//   D[m][0] accumulates the dot product for column colbase+m (f32 accumulate).
// w/alpha are read non-temporally (single use); hebb reads stay RT so the
// 67 MB hebb matrix remains resident in the 192 MB L2 for kernel 3's re-read.
// ---------------------------------------------------------------------------
__global__ void __launch_bounds__(256)
gemv_wmma_partials(const float* __restrict__ w,
                   const float* __restrict__ alpha,
                   const float* __restrict__ hebb,
                   const float* __restrict__ yin,
                   float* __restrict__ partial) {
    const int lane    = threadIdx.x & 31;
    const int wave    = threadIdx.x >> 5;
    const int m       = lane & 15;       // column within tile / row of A
    const int half    = lane >> 4;       // 0: K={0,1}, 1: K={2,3}
    const int colbase = (blockIdx.x * WAVES_PER_BLOCK + wave) * COLS_PER_WAVE;
    const int k0      = blockIdx.y * KCHUNK;

    v8f acc = {0.f, 0.f, 0.f, 0.f, 0.f, 0.f, 0.f, 0.f};

#pragma unroll 2
    for (int kb = k0; kb < k0 + KCHUNK; kb += 4) {
        const int r0 = kb + half * 2;
        const int i0 = r0 * N + colbase + m;   // < 2^24, int is safe
        const int i1 = i0 + N;

        // Form M = w + alpha*hebb for this lane's two A elements.
        const float a0 = __builtin_nontemporal_load(&w[i0]) +
                         __builtin_nontemporal_load(&alpha[i0]) * hebb[i0];
        const float a1 = __builtin_nontemporal_load(&w[i1]) +
                         __builtin_nontemporal_load(&alpha[i1]) * hebb[i1];

        // Wave-uniform yin loads (scalarizable), selected into lanes 0 / 16.
        const float s0 = yin[kb + 0];
        const float s1 = yin[kb + 1];
        const float s2 = yin[kb + 2];
        const float s3 = yin[kb + 3];

        v2f a = {a0, a1};
        v2f b = {(m == 0) ? (half ? s2 : s0) : 0.0f,
                 (m == 0) ? (half ? s3 : s1) : 0.0f};

        acc = __builtin_amdgcn_wmma_f32_16x16x4_f32(
            /*neg_a=*/false, a, /*neg_b=*/false, b,
            /*c_mod=*/(short)0, acc, /*reuse_a=*/false, /*reuse_b=*/false);
    }

    // D layout: VGPR g, lane 0 -> D[g][0] (col colbase+g);
    //           VGPR g, lane 16 -> D[g+8][0] (col colbase+8+g).
    if (m == 0) {
        float* dst = partial + blockIdx.y * N + colbase + half * 8;
#pragma unroll
        for (int g = 0; g < 8; ++g) dst[g] = acc[g];
    }
}

// ---------------------------------------------------------------------------
// Kernel 2: reduce KSPLIT partials per column, add input, tanh -> yout.
// ---------------------------------------------------------------------------
__global__ void __launch_bounds__(256)
finalize_yout(const float* __restrict__ partial,
              const float* __restrict__ input,
              float* __restrict__ yout) {
    const int j = blockIdx.x * blockDim.x + threadIdx.x;  // 4096 threads total
    float t = input[j];
#pragma unroll
    for (int p = 0; p < KSPLIT; ++p) t += partial[p * N + j];
    yout[j] = tanhf(t);
}

// ---------------------------------------------------------------------------
// Kernel 3: hebb' = (1-eta)*hebb + eta*yin[i]*yout[j], float4 streaming.
// hebb reads hit L2 (kept hot by kernel 1's TH policy); hebb' is streamed
// out with non-temporal stores (never re-read).
// ---------------------------------------------------------------------------
__global__ void __launch_bounds__(256)
hebb_update(const float* __restrict__ hebb,
            const float* __restrict__ yin,
            const float* __restrict__ yout,
            const float* __restrict__ eta_p,
            float* __restrict__ hebb_out) {
    const float eta = eta_p[0];
    const float om  = 1.0f - eta;

    const int idx = (blockIdx.x * blockDim.x + threadIdx.x) * 4;  // < 2^24
    const int row = idx >> 12;          // / N
    const int col = idx & (N - 1);      // % N

    const float ey = eta * yin[row];
    const v4f h  = *(const v4f*)(hebb + idx);
    const v4f yo = *(const v4f*)(yout + col);

    v4f r;
    r.x = om * h.x + ey * yo.x;
    r.y = om * h.y + ey * yo.y;
    r.z = om * h.z + ey * yo.z;
    r.w = om * h.w + ey * yo.w;
    __builtin_nontemporal_store(r, (v4f*)(hebb_out + idx));
}

// ---------------------------------------------------------------------------
extern "C" void kernel_launch(void* const* d_in, const int* in_sizes, int n_in,
                              void* d_out, int out_size, void* d_ws, size_t ws_size,
                              hipStream_t stream) {
    const float* input = (const float*)d_in[0];   // [1, N]
    const float* yin   = (const float*)d_in[1];   // [1, N]
    const float* hebb  = (const float*)d_in[2];   // [N, N]
    const float* w     = (const float*)d_in[3];   // [N, N]
    const float* alpha = (const float*)d_in[4];   // [N, N]
    const float* eta   = (const float*)d_in[5];   // [1]

    float* yout     = (float*)d_out;              // first N floats
    float* hebb_out = (float*)d_out + N;          // next N*N floats
    float* partial  = (float*)d_ws;               // KSPLIT * N floats (256 KB)

    // Phase 1: split-K WMMA GEMV partials. 32 x 16 blocks of 256 thr (4096 waves).
    gemv_wmma_partials<<<dim3(N / (COLS_PER_WAVE * WAVES_PER_BLOCK), KSPLIT),
                         256, 0, stream>>>(w, alpha, hebb, yin, partial);

    // Phase 2: reduce + tanh (4096 threads).
    finalize_yout<<<N / 256, 256, 0, stream>>>(partial, input, yout);

    // Phase 3: hebb update, float4 per thread: N*N/4 threads.
    hebb_update<<<(N * (N / 4)) / 256, 256, 0, stream>>>(hebb, yin, yout, eta,
                                                         hebb_out);
}